// SpikingRouter_53815940219182
// MI455X (gfx1250) — compile-verified
//
#include <hip/hip_runtime.h>

typedef __attribute__((ext_vector_type(2))) float v2f;
typedef __attribute__((ext_vector_type(4))) float v4f;
typedef __attribute__((ext_vector_type(8))) float v8f;

namespace {
constexpr int KDIM = 1024;   // feature dim
constexpr int EDIM = 64;     // experts
constexpr int KC   = 64;     // K chunk staged in LDS
constexpr int ROWS = 128;    // tokens per block (8 waves x 16 rows)
constexpr int PAD  = 68;     // LDS row stride in floats (272B: 16B-aligned, bank-skewed)
constexpr int NT   = 4;      // 4 expert tiles of 16
}

__global__ __launch_bounds__(256)
void spiking_router_wmma(const float* __restrict__ x,
                         const float* __restrict__ W,
                         const float* __restrict__ b,
                         float* __restrict__ routing,
                         float* __restrict__ logits_out,
                         int tokens)
{
    __shared__ float xs[ROWS * PAD];   // x tile; reused as logits tile after GEMM
    __shared__ float ws[EDIM * PAD];   // W tile
    __shared__ float thr[ROWS];        // per-token 2nd-max (top-2 threshold)

    const int tid     = threadIdx.x;
    const int lane    = tid & 31;
    const int wave    = tid >> 5;
    const int halfHi  = (lane >= 16) ? 1 : 0;
    const int r16     = lane & 15;
    const int waveRow = wave * 16;
    const long tokenBase = (long)blockIdx.x * ROWS;
    (void)tokens;

    v8f c[NT];
    #pragma unroll
    for (int n = 0; n < NT; ++n)
        #pragma unroll
        for (int j = 0; j < 8; ++j) c[n][j] = 0.0f;

    for (int kc = 0; kc < KDIM; kc += KC) {
        __syncthreads();
        // cooperative load x tile: 128 rows x 64 floats, float4-coalesced
        #pragma unroll
        for (int i = 0; i < 8; ++i) {
            int f   = tid + i * 256;
            int row = f >> 4;
            int c4  = (f & 15) * 4;
            *(v4f*)&xs[row * PAD + c4] =
                *(const v4f*)&x[(tokenBase + row) * KDIM + kc + c4];
        }
        // cooperative load W tile: 64 rows x 64 floats
        #pragma unroll
        for (int i = 0; i < 4; ++i) {
            int f   = tid + i * 256;
            int row = f >> 4;
            int c4  = (f & 15) * 4;
            *(v4f*)&ws[row * PAD + c4] =
                *(const v4f*)&W[(long)row * KDIM + kc + c4];
        }
        // prefetch next x chunk (global_prefetch_b8 path)
        if (kc + KC < KDIM) {
            __builtin_prefetch(
                &x[(tokenBase + (tid >> 1)) * KDIM + kc + KC + (tid & 1) * 32], 0, 1);
        }
        __syncthreads();

        // GEMM over this K chunk: 16 K-steps x 4 expert tiles of WMMA f32 16x16x4
        #pragma unroll 4
        for (int k2 = 0; k2 < KC; k2 += 4) {
            const int koff = k2 + (halfHi ? 2 : 0);
            // A frag: 16x4 fp32 (lanes 0-15: K=k2,k2+1; lanes 16-31: K=k2+2,k2+3)
            v2f a = *(const v2f*)&xs[(waveRow + r16) * PAD + koff];
            #pragma unroll
            for (int n = 0; n < NT; ++n) {
                // B frag: 4x16 = W[e][k] rows (B^T is W row-major)
                v2f bf = *(const v2f*)&ws[(n * 16 + r16) * PAD + koff];
                c[n] = __builtin_amdgcn_wmma_f32_16x16x4_f32(
                    false, a, false, bf, (short)0, c[n], false, false);
            }
        }
    }

    __syncthreads();   // all waves done reading xs; safe to reuse as logits tile

    // bias add + write logits (global + LDS)
    #pragma unroll
    for (int n = 0; n < NT; ++n) {
        float bias = b[n * 16 + r16];
        #pragma unroll
        for (int i = 0; i < 8; ++i) {
            float v = c[n][i] + bias;
            c[n][i] = v;
            int rl = waveRow + i + (halfHi ? 8 : 0);  // C/D layout: VGPR i = row i / i+8
            int e  = n * 16 + r16;
            xs[rl * PAD + e] = v;
            logits_out[(tokenBase + rl) * EDIM + e] = v;
        }
    }
    __syncthreads();

    // per-token 2nd-largest logit = top-2 membership threshold
    if (tid < ROWS) {
        float m1 = -3.4e38f, m2 = -3.4e38f;
        #pragma unroll
        for (int e = 0; e < EDIM; ++e) {
            float v = xs[tid * PAD + e];
            if (v > m1) { m2 = m1; m1 = v; }
            else if (v > m2) { m2 = v; }
        }
        thr[tid] = m2;
    }
    __syncthreads();

    // Scale-and-Fire quantize (step = 4.0/4 = 1.0) on top-2 slots, 0 elsewhere
    #pragma unroll
    for (int n = 0; n < NT; ++n) {
        #pragma unroll
        for (int i = 0; i < 8; ++i) {
            int rl = waveRow + i + (halfHi ? 8 : 0);
            int e  = n * 16 + r16;
            float v = c[n][i];
            float q = fminf(floorf(fmaxf(v, 0.0f)), 4.0f);
            routing[(tokenBase + rl) * EDIM + e] = (v >= thr[rl]) ? q : 0.0f;
        }
    }
}

extern "C" void kernel_launch(void* const* d_in, const int* in_sizes, int n_in,
                              void* d_out, int out_size, void* d_ws, size_t ws_size,
                              hipStream_t stream) {
    (void)n_in; (void)out_size; (void)d_ws; (void)ws_size;
    const float* x = (const float*)d_in[0];
    const float* W = (const float*)d_in[1];
    const float* b = (const float*)d_in[2];
    const int tokens = in_sizes[0] / KDIM;          // 8*4096 = 32768
    float* routing = (float*)d_out;                  // [tokens, 64]
    float* logits  = routing + (long)tokens * EDIM;  // [tokens, 64]
    dim3 grid(tokens / ROWS), block(256);
    hipLaunchKernelGGL(spiking_router_wmma, grid, block, 0, stream,
                       x, W, b, routing, logits, tokens);
}